// PointTransfomerEncModule_2680059592825
// MI455X (gfx1250) — compile-verified
//
#include <hip/hip_runtime.h>
#include <hip/hip_bf16.h>

// Problem constants (match reference)
#define BB   8
#define NN   8192
#define MM   2048
#define KK   16
#define CIN  64
#define COUT 128
#define CI   67      // 3 + CIN
#define CIP  68      // padded K-dim for WMMA (17 * 4)
#define BN_EPS 1e-5f

typedef float v2f __attribute__((ext_vector_type(2)));
typedef float v8f __attribute__((ext_vector_type(8)));

// ---------------------------------------------------------------------------
// Kernel 1: Furthest point sampling. One block per batch, 1024 threads,
// 8 points per thread held in registers. Sequential over M steps; argmax via
// wave32 shuffle reduce + 32-entry LDS tree.
// ---------------------------------------------------------------------------
__global__ __launch_bounds__(1024) void fps_kernel(
    const float* __restrict__ xyz, float* __restrict__ newXyz)
{
    const int b = blockIdx.x;
    const int t = threadIdx.x;
    const float* pts = xyz + (size_t)b * NN * 3;

    float px[8], py[8], pz[8], dd[8];
#pragma unroll
    for (int i = 0; i < 8; ++i) {
        int p = t + i * 1024;
        px[i] = pts[p * 3 + 0];
        py[i] = pts[p * 3 + 1];
        pz[i] = pts[p * 3 + 2];
        dd[i] = 1e10f;
    }

    __shared__ float sD[32];
    __shared__ int   sI[32];
    __shared__ float sCur[3];
    __shared__ int   sCurIdx;

    int   cur = 0;
    float cx = pts[0], cy = pts[1], cz = pts[2];

    for (int m = 0; m < MM; ++m) {
        if (t == 0) {
            newXyz[((size_t)b * MM + m) * 3 + 0] = cx;
            newXyz[((size_t)b * MM + m) * 3 + 1] = cy;
            newXyz[((size_t)b * MM + m) * 3 + 2] = cz;
        }
        float bestD = -1.0f;
        int   bestI = 0;
#pragma unroll
        for (int i = 0; i < 8; ++i) {
            float dx = px[i] - cx, dy = py[i] - cy, dz = pz[i] - cz;
            float d  = dx * dx + dy * dy + dz * dz;
            dd[i] = fminf(dd[i], d);
            int p = t + i * 1024;
            if (dd[i] > bestD) { bestD = dd[i]; bestI = p; }
        }
        // wave32 argmax (tie -> smaller index, matching jnp.argmax)
#pragma unroll
        for (int off = 16; off; off >>= 1) {
            float oD = __shfl_xor(bestD, off, 32);
            int   oI = __shfl_xor(bestI, off, 32);
            if (oD > bestD || (oD == bestD && oI < bestI)) { bestD = oD; bestI = oI; }
        }
        if ((t & 31) == 0) { sD[t >> 5] = bestD; sI[t >> 5] = bestI; }
        __syncthreads();
        if (t < 32) {
            bestD = sD[t]; bestI = sI[t];
#pragma unroll
            for (int off = 16; off; off >>= 1) {
                float oD = __shfl_xor(bestD, off, 32);
                int   oI = __shfl_xor(bestI, off, 32);
                if (oD > bestD || (oD == bestD && oI < bestI)) { bestD = oD; bestI = oI; }
            }
            if (t == 0) {
                sCurIdx = bestI;
                sCur[0] = pts[bestI * 3 + 0];
                sCur[1] = pts[bestI * 3 + 1];
                sCur[2] = pts[bestI * 3 + 2];
            }
        }
        __syncthreads();
        cur = sCurIdx;
        cx = sCur[0]; cy = sCur[1]; cz = sCur[2];
    }
    (void)cur;
}

// ---------------------------------------------------------------------------
// Kernel 2: brute-force KNN (top-16). One thread per query, points tiled
// through LDS. 'worst' register guard keeps the insertion path rare.
// ---------------------------------------------------------------------------
#define KT 1024
__global__ __launch_bounds__(256) void knn_kernel(
    const float* __restrict__ xyz, const float* __restrict__ newXyz,
    int* __restrict__ knnIdx)
{
    __shared__ float sP[KT * 3];
    const int b = blockIdx.x / (MM / 256);
    const int m = (blockIdx.x % (MM / 256)) * 256 + threadIdx.x;
    const float* pts = xyz + (size_t)b * NN * 3;

    const float tx = newXyz[((size_t)b * MM + m) * 3 + 0];
    const float ty = newXyz[((size_t)b * MM + m) * 3 + 1];
    const float tz = newXyz[((size_t)b * MM + m) * 3 + 2];

    float bd[KK];
    int   bi[KK];
#pragma unroll
    for (int i = 0; i < KK; ++i) { bd[i] = 3e38f; bi[i] = 0; }
    float worst = 3e38f;

    for (int tile = 0; tile < NN; tile += KT) {
        __syncthreads();
        for (int e = threadIdx.x; e < KT * 3; e += 256)
            sP[e] = pts[(size_t)tile * 3 + e];
        __syncthreads();
        for (int j = 0; j < KT; ++j) {
            float dx = sP[j * 3 + 0] - tx;
            float dy = sP[j * 3 + 1] - ty;
            float dz = sP[j * 3 + 2] - tz;
            float d  = dx * dx + dy * dy + dz * dz;
            if (d < worst) {
                int pos = KK - 1;
                while (pos > 0 && bd[pos - 1] > d) {
                    bd[pos] = bd[pos - 1]; bi[pos] = bi[pos - 1]; --pos;
                }
                bd[pos] = d; bi[pos] = tile + j;
                worst = bd[KK - 1];
            }
        }
    }
    int* o = knnIdx + ((size_t)b * MM + m) * KK;
#pragma unroll
    for (int i = 0; i < KK; ++i) o[i] = bi[i];
}

// ---------------------------------------------------------------------------
// Kernels 3 & 5: gather + WMMA GEMM (D = W[128x67] * grouped[67 x cols]).
// Block = 256 threads = 8 waves; handles 8 m's -> 128 columns gathered to LDS.
// Each wave: 16 columns (= one m's 16 neighbors) x all 128 channels
//   = 8 row-tiles x 17 k-steps of v_wmma_f32_16x16x4_f32.
// PASS 1: accumulate per-channel sum / sumsq for BN batch stats.
// PASS 2: apply BN scale/shift + ReLU, max-pool over the 16 neighbor columns
//         via 16-lane shuffle max, write pooled (B,COUT,M).
// ---------------------------------------------------------------------------
template <int PASS>
__global__ __launch_bounds__(256) void gemm_kernel(
    const float* __restrict__ xyz, const float* __restrict__ feat,
    const float* __restrict__ Wt, const int* __restrict__ knnIdx,
    float* __restrict__ sums, float* __restrict__ sqs,
    const float* __restrict__ scale, const float* __restrict__ shift,
    float* __restrict__ pooled)
{
    __shared__ float sG[128 * CIP];   // grouped columns, col-major [col][c]
    __shared__ int   sIdx[128];
    __shared__ float sSum[COUT];
    __shared__ float sSq[COUT];

    const int tid   = threadIdx.x;
    const int b     = blockIdx.x / (MM / 8);
    const int mBase = (blockIdx.x % (MM / 8)) * 8;

    if (tid < 128) {
        int mc = mBase + (tid >> 4);
        sIdx[tid] = knnIdx[((size_t)b * MM + mc) * KK + (tid & 15)];
        if constexpr (PASS == 1) { sSum[tid] = 0.f; sSq[tid] = 0.f; }
    }
    __syncthreads();

    // Gather 128 cols x 68 rows (row 67 zero-padded)
    for (int e = tid; e < 128 * CIP; e += 256) {
        int col = e / CIP;
        int c   = e - col * CIP;
        int p   = sIdx[col];
        float v;
        if (c < 3)        v = xyz[((size_t)b * NN + p) * 3 + c];
        else if (c < CI)  v = feat[((size_t)b * CIN + (c - 3)) * NN + p];
        else              v = 0.f;
        sG[col * CIP + c] = v;
    }
    __syncthreads();

    const int wave = tid >> 5;
    const int lane = tid & 31;
    const int hi   = lane >> 4;   // half-wave select
    const int ln   = lane & 15;
    const float* gcol = &sG[(wave * 16 + ln) * CIP];

    v8f acc[8];
#pragma unroll
    for (int r = 0; r < 8; ++r) acc[r] = {};

#pragma unroll
    for (int ks = 0; ks < 17; ++ks) {
        const int c0 = ks * 4 + 2 * hi;
        v2f bfrag;
        bfrag.x = gcol[c0];
        bfrag.y = gcol[c0 + 1];   // c0+1 == 67 hits the zero-pad row
#pragma unroll
        for (int r = 0; r < 8; ++r) {
            const int row = r * 16 + ln;
            v2f afrag;
            afrag.x = (c0 < CI)     ? Wt[row * CI + c0]     : 0.f;
            afrag.y = (c0 + 1 < CI) ? Wt[row * CI + c0 + 1] : 0.f;
            acc[r] = __builtin_amdgcn_wmma_f32_16x16x4_f32(
                false, afrag, false, bfrag, (short)0, acc[r], false, false);
        }
    }

    if constexpr (PASS == 1) {
#pragma unroll
        for (int r = 0; r < 8; ++r) {
#pragma unroll
            for (int j = 0; j < 8; ++j) {
                float v = acc[r][j];
                float s = v, q = v * v;
#pragma unroll
                for (int off = 8; off; off >>= 1) {   // stays within half-wave
                    s += __shfl_xor(s, off, 32);
                    q += __shfl_xor(q, off, 32);
                }
                if (ln == 0) {
                    int ch = r * 16 + j + 8 * hi;
                    atomicAdd(&sSum[ch], s);
                    atomicAdd(&sSq[ch], q);
                }
            }
        }
        __syncthreads();
        if (tid < COUT) {
            atomicAdd(&sums[tid], sSum[tid]);
            atomicAdd(&sqs[tid],  sSq[tid]);
        }
    } else {
        const int m = mBase + wave;   // this wave's 16 cols = one m's K nbrs
#pragma unroll
        for (int r = 0; r < 8; ++r) {
#pragma unroll
            for (int j = 0; j < 8; ++j) {
                int   ch = r * 16 + j + 8 * hi;
                float v  = fmaxf(acc[r][j] * scale[ch] + shift[ch], 0.f);
#pragma unroll
                for (int off = 8; off; off >>= 1)
                    v = fmaxf(v, __shfl_xor(v, off, 32));
                if (ln == 0)
                    pooled[((size_t)b * COUT + ch) * MM + m] = v;
            }
        }
    }
}

// ---------------------------------------------------------------------------
// Kernel 4: fold batch stats into BN scale/shift.
// ---------------------------------------------------------------------------
__global__ void finalize_kernel(
    const float* __restrict__ sums, const float* __restrict__ sqs,
    const float* __restrict__ gamma, const float* __restrict__ beta,
    float* __restrict__ scale, float* __restrict__ shift)
{
    int c = threadIdx.x;
    if (c >= COUT) return;
    const float inv  = 1.0f / (float)(BB * MM * KK);
    float mean = sums[c] * inv;
    float var  = sqs[c] * inv - mean * mean;
    float sc   = gamma[c] * rsqrtf(var + BN_EPS);
    scale[c] = sc;
    shift[c] = beta[c] - mean * sc;
}

// ---------------------------------------------------------------------------
extern "C" void kernel_launch(void* const* d_in, const int* in_sizes, int n_in,
                              void* d_out, int out_size, void* d_ws, size_t ws_size,
                              hipStream_t stream)
{
    (void)in_sizes; (void)n_in; (void)out_size; (void)ws_size;

    const float* xyz   = (const float*)d_in[0];   // (B,N,3)
    const float* feat  = (const float*)d_in[1];   // (B,CIN,N)
    const float* Wt    = (const float*)d_in[2];   // (COUT, 3+CIN)
    const float* gamma = (const float*)d_in[3];   // (COUT,)
    const float* beta  = (const float*)d_in[4];   // (COUT,)

    float* out    = (float*)d_out;
    float* newXyz = out;                          // B*M*3 floats
    float* pooled = out + (size_t)BB * MM * 3;    // B*COUT*M floats

    char*  ws     = (char*)d_ws;
    int*   knnIdx = (int*)ws;                                       // B*M*K ints
    float* sums   = (float*)(ws + (size_t)BB * MM * KK * 4);        // 128 f
    float* sqs    = sums + COUT;                                    // 128 f
    float* scale  = sqs + COUT;                                     // 128 f
    float* shift  = scale + COUT;                                   // 128 f

    hipMemsetAsync(sums, 0, 2 * COUT * sizeof(float), stream);

    fps_kernel<<<BB, 1024, 0, stream>>>(xyz, newXyz);
    knn_kernel<<<BB * (MM / 256), 256, 0, stream>>>(xyz, newXyz, knnIdx);
    gemm_kernel<1><<<BB * (MM / 8), 256, 0, stream>>>(
        xyz, feat, Wt, knnIdx, sums, sqs, nullptr, nullptr, nullptr);
    finalize_kernel<<<1, COUT, 0, stream>>>(sums, sqs, gamma, beta, scale, shift);
    gemm_kernel<2><<<BB * (MM / 8), 256, 0, stream>>>(
        xyz, feat, Wt, knnIdx, nullptr, nullptr, scale, shift, pooled);
}